// PNNLayer_83133386981888
// MI455X (gfx1250) — compile-verified
//
#include <hip/hip_runtime.h>

typedef __attribute__((ext_vector_type(2))) float v2f;
typedef __attribute__((ext_vector_type(8))) float v8f;

#define N_NODES 16384
#define N_ANCH  256
#define DLAT    32
#define BSTR    40   // padded LDS row stride in floats (2*40 % 64 == 16 -> conflict-free half-wave split)

// ---------------------------------------------------------------------------
// Prep 1: M'[a][j] = (1/A) * sum_d embeds[anchor_ids[a]][d] * W_hidden[d][j]
// grid = 256 blocks (one per anchor), block = 32 threads (one per out col)
// ---------------------------------------------------------------------------
__global__ void pnn_prep_M(const float* __restrict__ embeds,
                           const float* __restrict__ W,      // [2D][D] row-major
                           const int* __restrict__ anchor_ids,
                           float* __restrict__ Mp) {         // [A][D]
    const int a = blockIdx.x;
    const int j = threadIdx.x;
    const int aid = anchor_ids[a];
    const float* e = embeds + (size_t)aid * DLAT;
    float acc = 0.0f;
#pragma unroll
    for (int d = 0; d < DLAT; ++d)
        acc = fmaf(e[d], W[d * DLAT + j], acc);
    Mp[a * DLAT + j] = acc * (1.0f / (float)N_ANCH);
}

// ---------------------------------------------------------------------------
// Prep 2: C[j] = (1/A) * sum_{a<A} sum_d embeds[a][d] * W_hidden[D+d][j] + b[j]
// single block of 32 threads; trivial cost (8K FMA per lane)
// ---------------------------------------------------------------------------
__global__ void pnn_prep_C(const float* __restrict__ embeds,
                           const float* __restrict__ W,
                           const float* __restrict__ bias,
                           float* __restrict__ Cv) {          // [D]
    const int j = threadIdx.x;
    float acc = 0.0f;
    for (int a = 0; a < N_ANCH; ++a) {
        const float* e = embeds + (size_t)a * DLAT;
#pragma unroll
        for (int d = 0; d < DLAT; ++d)
            acc = fmaf(e[d], W[(DLAT + d) * DLAT + j], acc);
    }
    Cv[j] = acc * (1.0f / (float)N_ANCH) + bias[j];
}

// ---------------------------------------------------------------------------
// Main: out[N][32] = dists.T[N][A] @ M'[A][32] + C
// 256 threads = 8 waves per block; each wave computes a 16x32 output tile via
// V_WMMA_F32_16X16X4_F32 (fp32-exact; kernel is HBM-bound so f32 WMMA is the
// right matrix path on MI455X).
//
// f32 WMMA operand layouts (wave32):
//   A 16x4 : lane L holds (M = L%16, K = 2*(L/16) + {vgpr0,vgpr1})
//   B 4x16 : lane L holds (N = L%16, K = 2*(L/16) + {vgpr0,vgpr1})
//   C/D    : vgpr r -> (M = r + 8*(L/16), N = L%16)
// ---------------------------------------------------------------------------
__global__ __launch_bounds__(256) void pnn_main(
        const float* __restrict__ dists,  // [A][N] row-major
        const float* __restrict__ Mp,     // [A][32]
        const float* __restrict__ Cv,     // [32]
        float* __restrict__ out) {        // [N][32]
    __shared__ float Blds[N_ANCH * BSTR];

    const int tid = threadIdx.x;

    // Stage M' into LDS with padded stride (coalesced global read).
#pragma unroll 4
    for (int i = tid; i < N_ANCH * DLAT; i += 256) {
        const int a = i >> 5, j = i & 31;
        Blds[a * BSTR + j] = Mp[i];
    }
    __syncthreads();

    const int wave = tid >> 5;
    const int lane = tid & 31;
    const int m    = lane & 15;   // M (A-frag) / N (B,D-frag) index
    const int kh   = lane >> 4;   // half-wave: K offset selector
    const int node_base = (blockIdx.x * 8 + wave) * 16;

    const float c_lo = Cv[m];
    const float c_hi = Cv[m + 16];

    v8f acc0 = {};   // output cols 0..15
    v8f acc1 = {};   // output cols 16..31

    const float* aptr = dists + node_base + m;  // column (node) fixed per lane

#pragma unroll 4
    for (int k0 = 0; k0 < N_ANCH; k0 += 4) {
        const int ka = k0 + 2 * kh;
        v2f af, b0, b1;
        // A fragment: each half-wave reads 64B contiguous from one dists row.
        af.x = aptr[(size_t)ka * N_NODES];
        af.y = aptr[(size_t)(ka + 1) * N_NODES];
        // B fragments from LDS (bank-conflict-free via BSTR padding).
        b0.x = Blds[ka * BSTR + m];
        b0.y = Blds[(ka + 1) * BSTR + m];
        b1.x = Blds[ka * BSTR + m + 16];
        b1.y = Blds[(ka + 1) * BSTR + m + 16];

        acc0 = __builtin_amdgcn_wmma_f32_16x16x4_f32(
                   false, af, false, b0, (short)0, acc0, false, false);
        acc1 = __builtin_amdgcn_wmma_f32_16x16x4_f32(
                   false, af, false, b1, (short)0, acc1, false, false);
    }

    // Epilogue: add constant vector, store (each half-wave stores 64B rows).
#pragma unroll
    for (int r = 0; r < 8; ++r) {
        const int row = node_base + r + 8 * kh;
        out[(size_t)row * DLAT + m]      = acc0[r] + c_lo;
        out[(size_t)row * DLAT + m + 16] = acc1[r] + c_hi;
    }
}

// ---------------------------------------------------------------------------
extern "C" void kernel_launch(void* const* d_in, const int* in_sizes, int n_in,
                              void* d_out, int out_size, void* d_ws, size_t ws_size,
                              hipStream_t stream) {
    // setup_inputs order: embeds, dists, W_hidden, b_hidden, anchor_ids
    const float* embeds     = (const float*)d_in[0];
    const float* dists      = (const float*)d_in[1];
    const float* W_hidden   = (const float*)d_in[2];
    const float* b_hidden   = (const float*)d_in[3];
    const int*   anchor_ids = (const int*)d_in[4];

    float* Mp = (float*)d_ws;                 // [256][32]
    float* Cv = Mp + N_ANCH * DLAT;           // [32]
    float* out = (float*)d_out;               // [16384][32]

    pnn_prep_M<<<N_ANCH, DLAT, 0, stream>>>(embeds, W_hidden, anchor_ids, Mp);
    pnn_prep_C<<<1, DLAT, 0, stream>>>(embeds, W_hidden, b_hidden, Cv);

    // 1024 node tiles of 16 rows; 8 waves per block -> 128 blocks.
    pnn_main<<<N_NODES / (16 * 8), 256, 0, stream>>>(dists, Mp, Cv, out);
}